// Planner_49048526520802
// MI455X (gfx1250) — compile-verified
//
#include <hip/hip_runtime.h>

// Problem constants (from reference)
#define NCT   12          // N_CTPS
#define KDEG  4           // spline degree
#define ITERS 60          // OPT_ITERS
#define NSAMP 80          // N_SAMPLES
#define NTREE 64
#define LRc   0.1f
#define B1c   0.9f
#define B2c   0.999f
#define EPSc  1e-8f
#define THRESc 0.5f

typedef __attribute__((ext_vector_type(2))) float v2f;
typedef __attribute__((ext_vector_type(8))) float v8f;

// knots = [0,0,0,0, 0..8, 8,8,8,8]  ->  knot(i) = clamp(i-4, 0, 8)
__device__ __forceinline__ float knotv(int i) {
    int v = i - 4;
    v = v < 0 ? 0 : v;
    v = v > 8 ? 8 : v;
    return (float)v;
}

__global__ __launch_bounds__(32) void planner_kernel(
    const float* __restrict__ start_state,   // (B,13)
    const float* __restrict__ end_pos,       // (B,3)
    const float* __restrict__ trees,         // (B,64,2)
    float* __restrict__ out,                 // B*36 ctps floats, then B*17 knots
    int nb)
{
    const int b    = blockIdx.x;
    const int lane = threadIdx.x;

    __shared__ float s_basis[NSAMP][NCT];            // 80x12 basis
    __shared__ __align__(16) float s_trees[NTREE][2];
    __shared__ float s_traj[NSAMP][2];
    __shared__ float s_g[NSAMP][2];
    __shared__ float s_ctps[NCT][2];                 // current control points (x,y)
    __shared__ float s_cz[NCT];                      // z components (never change)

    __builtin_prefetch(trees + (size_t)b * NTREE * 2, 0, 0);  // global_prefetch_b8

    // ---- trees -> LDS via async DMA (no VGPR round trip; ASYNCcnt path) ----
    // 64 trees * 8B = 512B; each of the 32 lanes moves one 16B chunk.
    {
        unsigned lds_addr = (unsigned)(uintptr_t)(&s_trees[0][0]) + (unsigned)(lane * 16);
        unsigned long long gaddr =
            (unsigned long long)(const void*)(trees + (size_t)b * NTREE * 2) +
            (unsigned long long)(lane * 16);
        asm volatile("global_load_async_to_lds_b128 %0, %1, off"
                     :
                     : "v"(lds_addr), "v"(gaddr)
                     : "memory");
        asm volatile("s_wait_asynccnt 0x0" ::: "memory");
    }

    // ---- B-spline basis rows (Cox-de Boor, constant per dispatch) ----
    for (int ti = lane; ti < NSAMP; ti += 32) {
        float t = 8.0f * (float)ti / 79.0f;           // linspace(0, N_CTPS-K, 80)
        float Nv[16];
        const float last = knotv(16);
        #pragma unroll
        for (int j = 0; j < 16; ++j) {
            float left = knotv(j), right = knotv(j + 1);
            bool c1 = (t >= left) &&
                      ((t < right) || ((t >= last) && (right >= last) && (left < right)));
            Nv[j] = c1 ? 1.0f : 0.0f;
        }
        #pragma unroll
        for (int d = 1; d <= KDEG; ++d) {
            #pragma unroll
            for (int j = 0; j < 16; ++j) {
                if (j < 16 - d) {
                    float lo  = knotv(j),     hi1 = knotv(j + d);
                    float lo2 = knotv(j + 1), hi2 = knotv(j + d + 1);
                    float den1 = hi1 - lo, den2 = hi2 - lo2;
                    float w1 = (den1 > 0.0f) ? (t - lo)  / den1 : 0.0f;
                    float w2 = (den2 > 0.0f) ? (hi2 - t) / den2 : 0.0f;
                    Nv[j] = w1 * Nv[j] + w2 * Nv[j + 1];  // ascending in-place is safe
                }
            }
        }
        #pragma unroll
        for (int n = 0; n < NCT; ++n) s_basis[ti][n] = Nv[n];
    }

    // ---- ctps_init -> LDS (lane i owns control point i) ----
    if (lane < NCT) {
        const int i = lane;
        float step = (float)i / 11.0f;
        float sp[3], ep[3], c[3];
        #pragma unroll
        for (int k = 0; k < 3; ++k) {
            sp[k] = start_state[(size_t)b * 13 + k];
            ep[k] = end_pos[(size_t)b * 3 + k];
            c[k]  = sp[k] + step * (ep[k] - sp[k]);
        }
        if (i == 1) {
            #pragma unroll
            for (int k = 0; k < 3; ++k)
                c[k] = sp[k] + start_state[(size_t)b * 13 + 7 + k] * (1.0f / (float)KDEG);
        }
        s_ctps[i][0] = c[0];
        s_ctps[i][1] = c[1];
        s_cz[i]      = c[2];
    }
    __syncthreads();

    // ---- Precompute loop-invariant WMMA A-fragments (basis, f32 16x16x4) ----
    // A 16x4 layout: lanes 0-15 -> M=row, {K0,K1}; lanes 16-31 -> M=row, {K2,K3}
    const int rowA = lane & 15;
    const int kkA  = (lane >> 4) << 1;        // 0 or 2
    v2f aF[5][3];
    #pragma unroll
    for (int mt = 0; mt < 5; ++mt) {
        #pragma unroll
        for (int kc = 0; kc < 3; ++kc) {
            int m = mt * 16 + rowA;
            float a0 = 0.0f, a1 = 0.0f;
            if (m < NSAMP) {
                a0 = s_basis[m][kc * 4 + kkA];
                a1 = s_basis[m][kc * 4 + kkA + 1];
            }
            aF[mt][kc].x = a0;
            aF[mt][kc].y = a1;
        }
    }

    // B 4x16 layout: lanes 0-15 -> N=lane, rows {K0,K1}; lanes 16-31 -> N=lane-16, rows {K2,K3}
    const int dN = (lane < 16) ? lane : (lane - 16);   // output column (0=x, 1=y used)
    const int rB = (lane < 16) ? 0 : 2;

    // Adam state: lane q (0..11) owns free ctp n=q/2, dim d=q%2
    const int an = lane >> 1, ad = lane & 1;
    float xval = 0.0f, mAd = 0.0f, vAd = 0.0f;
    if (lane < 12) xval = s_ctps[3 + an][ad];

    float b1p = 1.0f, b2p = 1.0f;

    for (int it = 0; it < ITERS; ++it) {
        b1p *= B1c;
        b2p *= B2c;
        __syncthreads();   // s_ctps from previous Adam update visible

        // ---- B fragments from current control points ----
        v2f bF[3];
        #pragma unroll
        for (int kc = 0; kc < 3; ++kc) {
            float bx = 0.0f, by = 0.0f;
            if (dN < 2) {
                bx = s_ctps[kc * 4 + rB + 0][dN];
                by = s_ctps[kc * 4 + rB + 1][dN];
            }
            bF[kc].x = bx;
            bF[kc].y = by;
        }

        // ---- traj = basis @ ctps_xy via V_WMMA_F32_16X16X4_F32 ----
        #pragma unroll
        for (int mt = 0; mt < 5; ++mt) {
            v8f acc = {0.f, 0.f, 0.f, 0.f, 0.f, 0.f, 0.f, 0.f};
            #pragma unroll
            for (int kc = 0; kc < 3; ++kc) {
                acc = __builtin_amdgcn_wmma_f32_16x16x4_f32(
                    false, aF[mt][kc], false, bF[kc], (short)0, acc, false, false);
            }
            // C/D layout: VGPR r -> (M=r, N=lane) lanes 0-15 ; (M=r+8, N=lane-16) lanes 16-31
            if (dN < 2) {
                #pragma unroll
                for (int r = 0; r < 8; ++r) {
                    int m = mt * 16 + ((lane < 16) ? r : (r + 8));
                    if (m < NSAMP) s_traj[m][dN] = acc[r];
                }
            }
        }
        __syncthreads();

        // ---- min-distance hinge gradient per waypoint ----
        for (int t = lane; t < NSAMP; t += 32) {
            float px = s_traj[t][0], py = s_traj[t][1];
            float best = 3.4e38f, btx = 0.0f, bty = 0.0f;
            for (int j = 0; j < NTREE; ++j) {
                float tx = s_trees[j][0], ty = s_trees[j][1];
                float dx = px - tx, dy = py - ty;
                float d2 = dx * dx + dy * dy;
                if (d2 < best) { best = d2; btx = tx; bty = ty; }
            }
            float dist = sqrtf(fmaxf(best, 1e-12f));
            float gx = 0.0f, gy = 0.0f;
            if (dist < THRESc) {          // d(THRES - mind)/d wpt = -(wpt - tree)/dist
                float inv = 1.0f / dist;
                gx = -(px - btx) * inv;
                gy = -(py - bty) * inv;
            }
            s_g[t][0] = gx;
            s_g[t][1] = gy;
        }
        __syncthreads();

        // ---- project gradient onto free control points: g_x = basis[:,3:9]^T @ g ----
        float part[12];
        #pragma unroll
        for (int q = 0; q < 12; ++q) part[q] = 0.0f;
        for (int t = lane; t < NSAMP; t += 32) {
            float g0 = s_g[t][0], g1 = s_g[t][1];
            #pragma unroll
            for (int n = 0; n < 6; ++n) {
                float bb = s_basis[t][3 + n];
                part[2 * n]     += bb * g0;
                part[2 * n + 1] += bb * g1;
            }
        }
        #pragma unroll
        for (int q = 0; q < 12; ++q) {
            float v = part[q];
            #pragma unroll
            for (int off = 16; off > 0; off >>= 1)
                v += __shfl_xor(v, off, 32);
            part[q] = v;
        }

        // ---- Adam update (lanes 0..11, one scalar each) ----
        if (lane < 12) {
            float g = 0.0f;
            #pragma unroll
            for (int q = 0; q < 12; ++q)
                if (lane == q) g = part[q];
            mAd = B1c * mAd + (1.0f - B1c) * g;
            vAd = B2c * vAd + (1.0f - B2c) * g * g;
            float mhat = mAd / (1.0f - b1p);
            float vhat = vAd / (1.0f - b2p);
            xval = xval - LRc * mhat / (sqrtf(vhat) + EPSc);
            s_ctps[3 + an][ad] = xval;
        }
    }
    __syncthreads();

    // ---- write outputs: ctps (B,1,12,3) then knots (B,1,17) ----
    for (int q = lane; q < 36; q += 32) {
        int i = q / 3, d = q - i * 3;
        float val = (d < 2) ? s_ctps[i][d] : s_cz[i];
        out[(size_t)b * 36 + q] = val;
    }
    if (lane < 17) {
        out[(size_t)nb * 36 + (size_t)b * 17 + lane] = knotv(lane);
    }
}

extern "C" void kernel_launch(void* const* d_in, const int* in_sizes, int n_in,
                              void* d_out, int out_size, void* d_ws, size_t ws_size,
                              hipStream_t stream) {
    const float* start_state = (const float*)d_in[0];
    const float* end_pos     = (const float*)d_in[1];
    const float* trees       = (const float*)d_in[2];
    float* out = (float*)d_out;
    const int nb = in_sizes[1] / 3;   // 512 batch items
    planner_kernel<<<nb, 32, 0, stream>>>(start_state, end_pos, trees, out, nb);
}